// MultiHeadAttention_62912680952574
// MI455X (gfx1250) — compile-verified
//
#include <hip/hip_runtime.h>
#include <hip/hip_bf16.h>

typedef __attribute__((ext_vector_type(2))) float v2f;
typedef __attribute__((ext_vector_type(4))) float v4f;
typedef __attribute__((ext_vector_type(8))) float v8f;

#define EMBED 512
#define QKV 64
#define HEADS 8
#define HEAD_DIM 8
#define BATCH 8
#define SEQ 2048
#define NTOK (BATCH * SEQ)   /* 16384 */
#define NGRP (BATCH * HEADS) /* 64 */
#define RSQRT_HD 0.35355339059327373f

// D = A(16x4 f32) * B(4x16 f32) + C(16x16 f32), exact fp32 on the matrix pipe.
__device__ __forceinline__ v8f wmma4(v2f a, v2f b, v8f c) {
  return __builtin_amdgcn_wmma_f32_16x16x4_f32(
      /*neg_a=*/false, a, /*neg_b=*/false, b,
      /*c_mod=*/(short)0, c, /*reuse_a=*/false, /*reuse_b=*/false);
}

// LDS byte offset of a generic pointer to __shared__ (ISA 10.2: offset = low 32 bits of the address).
__device__ __forceinline__ uint32_t lds_off(const float* p) {
  return (uint32_t)(uintptr_t)p;
}

// CDNA5 async copies: global -> LDS, per-lane addresses both sides, ASYNCcnt.
__device__ __forceinline__ void async_copy_b64(uint32_t lds, const float* g) {
  asm volatile("global_load_async_to_lds_b64 %0, %1, off" ::"v"(lds), "v"(g)
               : "memory");
}
__device__ __forceinline__ void async_copy_b32(uint32_t lds, const float* g) {
  asm volatile("global_load_async_to_lds_b32 %0, %1, off" ::"v"(lds), "v"(g)
               : "memory");
}
__device__ __forceinline__ void wait_async0() {
  asm volatile("s_wait_asynccnt 0" ::: "memory");
}

// ---------------------------------------------------------------------------
// Kernel 1: fused Q/K/V projection.  [16384,512] x [512,64] (+bias) x3
// ---------------------------------------------------------------------------
__global__ __launch_bounds__(256) void qkv_proj_kernel(
    const float* __restrict__ x,
    const float* __restrict__ Wq, const float* __restrict__ bq,
    const float* __restrict__ Wk, const float* __restrict__ bk,
    const float* __restrict__ Wv, const float* __restrict__ bv,
    float* __restrict__ Qb, float* __restrict__ Kb, float* __restrict__ Vb) {
  __shared__ float Xs[32 * 68];
  __shared__ float Ws0[64 * 64];
  __shared__ float Ws1[64 * 64];
  __shared__ float Ws2[64 * 64];

  const int t    = threadIdx.x;
  const int wave = t >> 5;
  const int lane = t & 31;
  const int hl   = lane >> 4;
  const int ln   = lane & 15;
  const int mbase = blockIdx.x * 32;
  const int wr = wave >> 2, wc = wave & 3;
  const int nb = wc * 16;

  v8f accq = {}; v8f acck = {}; v8f accv = {};

  for (int kc = 0; kc < EMBED; kc += 64) {
    __syncthreads();
#pragma unroll
    for (int i = 0; i < 2; ++i) {
      int idx4 = i * 256 + t;
      int row = idx4 >> 4, col = (idx4 & 15) << 2;
      *(v4f*)(&Xs[row * 68 + col]) =
          *(const v4f*)(x + (size_t)(mbase + row) * EMBED + kc + col);
    }
#pragma unroll
    for (int i = 0; i < 4; ++i) {
      int idx4 = i * 256 + t;
      int row = idx4 >> 4, col = (idx4 & 15) << 2;
      size_t g = (size_t)(kc + row) * QKV + col;
      *(v4f*)(&Ws0[row * 64 + col]) = *(const v4f*)(Wq + g);
      *(v4f*)(&Ws1[row * 64 + col]) = *(const v4f*)(Wk + g);
      *(v4f*)(&Ws2[row * 64 + col]) = *(const v4f*)(Wv + g);
    }
    __syncthreads();

    const int arow = (wr * 16 + ln) * 68;
#pragma unroll
    for (int c = 0; c < 16; ++c) {
      const int kk = 4 * c + 2 * hl;
      v2f a = *(const v2f*)(&Xs[arow + kk]);
      v2f b;
      b.x = Ws0[kk * 64 + nb + ln]; b.y = Ws0[(kk + 1) * 64 + nb + ln];
      accq = wmma4(a, b, accq);
      b.x = Ws1[kk * 64 + nb + ln]; b.y = Ws1[(kk + 1) * 64 + nb + ln];
      acck = wmma4(a, b, acck);
      b.x = Ws2[kk * 64 + nb + ln]; b.y = Ws2[(kk + 1) * 64 + nb + ln];
      accv = wmma4(a, b, accv);
    }
  }

  const int colg = nb + ln;
  const float biasq = bq[colg], biask = bk[colg], biasv = bv[colg];
#pragma unroll
  for (int r = 0; r < 8; ++r) {
    size_t rowg = (size_t)(mbase + wr * 16 + r + 8 * hl);
    Qb[rowg * QKV + colg] = accq[r] + biasq;
    Kb[rowg * QKV + colg] = acck[r] + biask;
    Vb[rowg * QKV + colg] = accv[r] + biasv;
  }
}

// ---------------------------------------------------------------------------
// Kernel 2: flash attention. Q/K/V viewed as [NGRP*SEQ, 8].
// grid = (64 groups, 16 q-blocks); 8 waves x 16 queries. Double-buffered
// 64-key chunks streamed global->LDS with async copies (V transposed in
// flight). Softmax denominator = "ones column" (col 8) of the P*V WMMA.
// ---------------------------------------------------------------------------
__global__ __launch_bounds__(256) void flash_attn_kernel(
    const float* __restrict__ Qb, const float* __restrict__ Kb,
    const float* __restrict__ Vb, const float* __restrict__ mask,
    float* __restrict__ Ob) {
  __shared__ float Klds[2][64 * 10];  // [key][dim], stride 10
  __shared__ float VT[2][9 * 72];     // [dim][key], stride 72; row 8 = ones
  __shared__ float Plds[8 * 16 * 18]; // per-wave P staging (D->A relayout)

  const int t    = threadIdx.x;
  const int wave = t >> 5;
  const int lane = t & 31;
  const int hl   = lane >> 4;
  const int ln   = lane & 15;
  const int g    = blockIdx.x;
  const int qblk = blockIdx.y * 128;
  const int qloc = qblk + wave * 16 + ln;

  // Q tile in A layout: lane = row; VGPR pair = K dims {2hl,2hl+1}/{4+2hl,..}
  const float* Qrow = Qb + ((size_t)g * SEQ + qloc) * HEAD_DIM;
  const v2f aq0 = *(const v2f*)(Qrow + 2 * hl);
  const v2f aq1 = *(const v2f*)(Qrow + 4 + 2 * hl);

  // ones column for the free softmax denominator (never overwritten)
  if (t < 64) { VT[0][8 * 72 + t] = 1.0f; VT[1][8 * 72 + t] = 1.0f; }

  // async-copy one 64-key x 8-dim chunk: K row-major, V transposed (scatter)
  auto issue_chunk = [&](int buf, int kc) {
    const float* ks = Kb + ((size_t)g * SEQ + kc) * HEAD_DIM;
    const float* vs = Vb + ((size_t)g * SEQ + kc) * HEAD_DIM;
    const int i0 = 2 * t;            // two dwords per thread, same key row
    async_copy_b64(lds_off(&Klds[buf][(i0 >> 3) * 10 + (i0 & 7)]), ks + i0);
    async_copy_b32(lds_off(&VT[buf][(i0 & 7) * 72 + (i0 >> 3)]), vs + i0);
    const int i1 = i0 + 1;
    async_copy_b32(lds_off(&VT[buf][(i1 & 7) * 72 + (i1 >> 3)]), vs + i1);
  };

  issue_chunk(0, 0);

  v8f acc = {};
  float mrow[8];
#pragma unroll
  for (int r = 0; r < 8; ++r) mrow[r] = -3.0e38f;
  const int pbase = wave * (16 * 18);
  const int vdim  = (ln < 9) ? ln : 8;   // cols 9..15 ride the ones row

  wait_async0();
  __syncthreads();

  for (int it = 0; it < SEQ / 64; ++it) {
    const int cur = it & 1;
    const int kc  = it * 64;
    if (it + 1 < SEQ / 64) issue_chunk(1 - cur, kc + 64);  // overlap next chunk

    const float* Kc = Klds[cur];
    const float* Vc = VT[cur];
#pragma unroll
    for (int kt = 0; kt < 4; ++kt) {
      const int kb = kt * 16;
      // ---- S = Q K^T (two K=4 steps over head_dim) ----
      v8f s = {};
      const int krow = (kb + ln) * 10;
      v2f b0 = *(const v2f*)(&Kc[krow + 2 * hl]);
      s = wmma4(aq0, b0, s);
      v2f b1 = *(const v2f*)(&Kc[krow + 4 + 2 * hl]);
      s = wmma4(aq1, b1, s);

      // ---- mask, scale, online max; P -> LDS in A layout ----
#pragma unroll
      for (int r = 0; r < 8; ++r) {
        const int ql = qblk + wave * 16 + r + 8 * hl;
        const float mv = mask[(size_t)ql * SEQ + kc + kb + ln];
        const float sv = (s[r] + mv) * RSQRT_HD;
        float tm = sv;
        tm = fmaxf(tm, __shfl_xor(tm, 1));
        tm = fmaxf(tm, __shfl_xor(tm, 2));
        tm = fmaxf(tm, __shfl_xor(tm, 4));
        tm = fmaxf(tm, __shfl_xor(tm, 8));
        const float mnew  = fmaxf(mrow[r], tm);
        const float scale = __expf(mrow[r] - mnew);
        mrow[r] = mnew;
        acc[r] *= scale;                   // rescales O *and* the denom column
        Plds[pbase + (r + 8 * hl) * 18 + ln] = __expf(sv - mnew);
      }

      // ---- O(+denominator col) += P V  (four K=4 steps over 16 keys) ----
#pragma unroll
      for (int j = 0; j < 4; ++j) {
        v2f pa = *(const v2f*)(&Plds[pbase + ln * 18 + 4 * j + 2 * hl]);
        v2f vb = *(const v2f*)(&Vc[vdim * 72 + kb + 4 * j + 2 * hl]);
        acc = wmma4(pa, vb, acc);
      }
    }

    if (it + 1 < SEQ / 64) wait_async0();  // own async writes landed
    __syncthreads();                       // everyone's writes visible
  }

  // divide by the denominator broadcast from column 8 (lanes 8 / 24)
#pragma unroll
  for (int r = 0; r < 8; ++r) {
    const float l = __shfl(acc[r], (lane & 16) | 8);
    acc[r] = acc[r] / l;
  }
  if (ln < HEAD_DIM) {
#pragma unroll
    for (int r = 0; r < 8; ++r) {
      const int ql = qblk + wave * 16 + r + 8 * hl;
      Ob[((size_t)g * SEQ + ql) * HEAD_DIM + ln] = acc[r];
    }
  }
}

// ---------------------------------------------------------------------------
// Kernel 3: output projection.  [16384,64] x [64,512] + bias.
// ---------------------------------------------------------------------------
__global__ __launch_bounds__(256) void out_proj_kernel(
    const float* __restrict__ A, const float* __restrict__ Wu,
    const float* __restrict__ bu, float* __restrict__ out) {
  __shared__ float As[32 * 68];
  __shared__ float Bs[64 * 128];

  const int t    = threadIdx.x;
  const int wave = t >> 5;
  const int lane = t & 31;
  const int hl   = lane >> 4;
  const int ln   = lane & 15;
  const int mbase = blockIdx.x * 32;
  const int nbase = blockIdx.y * 128;
  const int wr = wave >> 2, wc = wave & 3;

#pragma unroll
  for (int i = 0; i < 2; ++i) {
    int idx4 = i * 256 + t;
    int row = idx4 >> 4, col = (idx4 & 15) << 2;
    *(v4f*)(&As[row * 68 + col]) =
        *(const v4f*)(A + (size_t)(mbase + row) * QKV + col);
  }
#pragma unroll
  for (int i = 0; i < 8; ++i) {
    int idx4 = i * 256 + t;
    int row = idx4 >> 5, col = (idx4 & 31) << 2;
    *(v4f*)(&Bs[row * 128 + col]) =
        *(const v4f*)(Wu + (size_t)row * EMBED + nbase + col);
  }
  __syncthreads();

  v8f acc0 = {}; v8f acc1 = {};
  const int arow = (wr * 16 + ln) * 68;
  const int nb0 = wc * 32, nb1 = wc * 32 + 16;
#pragma unroll
  for (int c = 0; c < 16; ++c) {
    const int kk = 4 * c + 2 * hl;
    v2f a = *(const v2f*)(&As[arow + kk]);
    v2f b;
    b.x = Bs[kk * 128 + nb0 + ln]; b.y = Bs[(kk + 1) * 128 + nb0 + ln];
    acc0 = wmma4(a, b, acc0);
    b.x = Bs[kk * 128 + nb1 + ln]; b.y = Bs[(kk + 1) * 128 + nb1 + ln];
    acc1 = wmma4(a, b, acc1);
  }

  const float bias0 = bu[nbase + nb0 + ln];
  const float bias1 = bu[nbase + nb1 + ln];
#pragma unroll
  for (int r = 0; r < 8; ++r) {
    size_t row = (size_t)(mbase + wr * 16 + r + 8 * hl);
    out[row * EMBED + nbase + nb0 + ln] = acc0[r] + bias0;
    out[row * EMBED + nbase + nb1 + ln] = acc1[r] + bias1;
  }
}

// ---------------------------------------------------------------------------
extern "C" void kernel_launch(void* const* d_in, const int* in_sizes, int n_in,
                              void* d_out, int out_size, void* d_ws, size_t ws_size,
                              hipStream_t stream) {
  const float* x    = (const float*)d_in[0];
  const float* mask = (const float*)d_in[1];
  const float* Wq   = (const float*)d_in[2];
  const float* bq   = (const float*)d_in[3];
  const float* Wk   = (const float*)d_in[4];
  const float* bk   = (const float*)d_in[5];
  const float* Wv   = (const float*)d_in[6];
  const float* bv   = (const float*)d_in[7];
  const float* Wu   = (const float*)d_in[8];
  const float* bu   = (const float*)d_in[9];
  float* out = (float*)d_out;

  // workspace: Q | K | V | attn, each NTOK*QKV floats (4 MB) -> 16 MB total
  const size_t n = (size_t)NTOK * QKV;
  float* Qb = (float*)d_ws;
  float* Kb = Qb + n;
  float* Vb = Kb + n;
  float* Ob = Vb + n;

  qkv_proj_kernel<<<dim3(NTOK / 32), dim3(256), 0, stream>>>(
      x, Wq, bq, Wk, bk, Wv, bv, Qb, Kb, Vb);
  flash_attn_kernel<<<dim3(NGRP, SEQ / 128), dim3(256), 0, stream>>>(
      Qb, Kb, Vb, mask, Ob);
  out_proj_kernel<<<dim3(NTOK / 32, EMBED / 128), dim3(256), 0, stream>>>(
      Ob, Wu, bu, out);
}